// SlaterWF_87110526698142
// MI455X (gfx1250) — compile-verified
//
#include <hip/hip_runtime.h>
#include <hip/hip_bf16.h>

// ---------------------------------------------------------------------------
// Slater wavefunction: per sample s
//   S_up = aos[s,0:16,:]  @ (ovlps[:,None] * mo[:,occ_up])   (16x128 @ 128x16)
//   S_dn = aos[s,16:32,:] @ (ovlps[:,None] * mo[:,occ_dn])
//   out[s] = (1/16!) * det(S_up) * det(S_dn)
// Memory-bound (819 MB aos stream @ 23.3 TB/s -> ~35us floor), so stay f32
// and use V_WMMA_F32_16X16X4_F32 chained over K (one wave per sample).
// ---------------------------------------------------------------------------

typedef float v2f __attribute__((ext_vector_type(2)));
typedef float v8f __attribute__((ext_vector_type(8)));

#define N_SAMPLES 50000
#define N_ELEC    32
#define N_BASIS   128
#define N_OCC     16
#define WAVES_PER_BLOCK 8
#define BLOCK_THREADS   (WAVES_PER_BLOCK * 32)
#define DET_STRIDE 17   // pad 16x16 det scratch to kill LDS bank conflicts

__global__ __launch_bounds__(BLOCK_THREADS)
void slater_det_kernel(const float* __restrict__ aos,
                       const float* __restrict__ ovlps,
                       const float* __restrict__ mo,
                       const int*   __restrict__ occ_up,
                       const int*   __restrict__ occ_dn,
                       float*       __restrict__ out)
{
    // Scaled + gathered coefficient matrices, [k][o] layout, o contiguous.
    __shared__ float Bs_up[N_BASIS * N_OCC];
    __shared__ float Bs_dn[N_BASIS * N_OCC];
    // Per-wave determinant scratch: up matrix then down matrix, column-major,
    // padded stride.
    __shared__ float Sdet[WAVES_PER_BLOCK][2 * 16 * DET_STRIDE];

    const int tid = threadIdx.x;

    // Cooperative build of B = diag(ovlps) * mo[:, occ] (2048 elems per matrix).
    for (int i = tid; i < N_BASIS * N_OCC; i += BLOCK_THREADS) {
        const int k = i >> 4;   // basis index
        const int o = i & 15;   // occupied-orbital slot
        const float ov = ovlps[k];
        Bs_up[i] = ov * mo[k * N_BASIS + occ_up[o]];
        Bs_dn[i] = ov * mo[k * N_BASIS + occ_dn[o]];
    }
    __syncthreads();

    const int wave   = tid >> 5;
    const int lane   = tid & 31;
    const int sample = blockIdx.x * WAVES_PER_BLOCK + wave;
    if (sample >= N_SAMPLES) return;   // wave-uniform: EXEC stays all-1s

    // f32 16x16x4 WMMA operand layout (ISA 7.12.2):
    //   A (16x4): lane l -> row = l%16; VGPR0 holds K = 2*(l/16), VGPR1 K+1
    //   B (4x16): lane l -> col = l%16; VGPR0 holds K = 2*(l/16), VGPR1 K+1
    //   C/D     : lane l -> col = l%16; VGPR v holds row v + 8*(l/16)
    const int row = lane & 15;
    const int hi  = lane >> 4;
    const int col = lane & 15;

    const float* Aup = aos + (size_t)sample * (N_ELEC * N_BASIS)
                           + row * N_BASIS + 2 * hi;
    const float* Adn = Aup + N_OCC * N_BASIS;

    v8f acc_up = {0.f, 0.f, 0.f, 0.f, 0.f, 0.f, 0.f, 0.f};
    v8f acc_dn = {0.f, 0.f, 0.f, 0.f, 0.f, 0.f, 0.f, 0.f};

    // K = 128 in 32 steps of 4; two independent accumulator chains (up/dn)
    // interleave to hide WMMA latency. A loads are 8B/lane global b64 loads.
    #pragma unroll
    for (int t = 0; t < N_BASIS / 4; ++t) {
        const int k0 = 4 * t;
        const v2f a_up = *(const v2f*)(Aup + k0);
        const v2f a_dn = *(const v2f*)(Adn + k0);
        v2f b_up, b_dn;
        b_up.x = Bs_up[(k0 + 2 * hi    ) * N_OCC + col];
        b_up.y = Bs_up[(k0 + 2 * hi + 1) * N_OCC + col];
        b_dn.x = Bs_dn[(k0 + 2 * hi    ) * N_OCC + col];
        b_dn.y = Bs_dn[(k0 + 2 * hi + 1) * N_OCC + col];
        acc_up = __builtin_amdgcn_wmma_f32_16x16x4_f32(
            false, a_up, false, b_up, (short)0, acc_up, false, false);
        acc_dn = __builtin_amdgcn_wmma_f32_16x16x4_f32(
            false, a_dn, false, b_dn, (short)0, acc_dn, false, false);
    }

    // Spill both 16x16 results to LDS column-major (Sdet[mat][col][row]).
    float* Su = &Sdet[wave][0];
    float* Sd = &Sdet[wave][16 * DET_STRIDE];
    #pragma unroll
    for (int v = 0; v < 8; ++v) {
        Su[col * DET_STRIDE + 8 * hi + v] = acc_up[v];
        Sd[col * DET_STRIDE + 8 * hi + v] = acc_dn[v];
    }
    // Same-wave LDS produce->consume: hardware DScnt waits inserted by compiler.

    // Cooperative Gaussian elimination, one column per lane (det(S)=det(S^T)).
    // Lanes 0..15 reduce S_up, lanes 16..31 reduce S_dn; __shfl with width=16
    // broadcasts within each half of the wave32 (lowered to lane permutes).
    const float* Scol = (lane < 16) ? (Su + lane * DET_STRIDE)
                                    : (Sd + (lane - 16) * DET_STRIDE);
    float m[16];
    #pragma unroll
    for (int i = 0; i < 16; ++i) m[i] = Scol[i];

    float det = 1.0f;
    const int c = lane & 15;
    #pragma unroll
    for (int k = 0; k < 16; ++k) {
        const float p = __shfl(m[k], k, 16);          // pivot M[k][k]
        det *= p;
        // v_rcp_f32 (~1 ULP) instead of the ~7-instruction exact-division
        // expansion: the elimination factor tolerates it, and it shortens the
        // serial VALU tail that competes with the memory roofline.
        const float pinv = __builtin_amdgcn_rcpf(p);
        const float f = (c > k) ? (m[k] * pinv) : 0.0f;  // branchless
        #pragma unroll
        for (int i = k + 1; i < 16; ++i) {
            const float mki = __shfl(m[i], k, 16);    // pivot row element
            m[i] = fmaf(-f, mki, m[i]);
        }
    }

    // Combine halves: lane l pairs with lane l^16 (other determinant).
    const float other = __shfl_xor(det, 16, 32);
    const float inv_fact16 = 4.779477332387385e-14f;  // 1/16! = norm_up*norm_dn
    const float result = det * other * inv_fact16;
    if (lane == 0) out[sample] = result;
}

extern "C" void kernel_launch(void* const* d_in, const int* in_sizes, int n_in,
                              void* d_out, int out_size, void* d_ws, size_t ws_size,
                              hipStream_t stream) {
    (void)in_sizes; (void)n_in; (void)d_ws; (void)ws_size; (void)out_size;
    const float* aos    = (const float*)d_in[0];
    const float* ovlps  = (const float*)d_in[1];
    const float* mo     = (const float*)d_in[2];
    const int*   occ_up = (const int*)d_in[3];
    const int*   occ_dn = (const int*)d_in[4];
    // d_in[5] = n_up (scalar) — fixed at 16 in this problem size.
    float* out = (float*)d_out;

    const int blocks = (N_SAMPLES + WAVES_PER_BLOCK - 1) / WAVES_PER_BLOCK;
    slater_det_kernel<<<dim3(blocks), dim3(BLOCK_THREADS), 0, stream>>>(
        aos, ovlps, mo, occ_up, occ_dn, out);
}